// SharedEncoder_20143396618416
// MI455X (gfx1250) — compile-verified
//
#include <hip/hip_runtime.h>
#include <hip/hip_bf16.h>
#include <math.h>

// ---------------------------------------------------------------------------
// 2-layer GAT (PyG-style) for gfx1250 / MI455X.
// Dense h@W GEMMs use V_WMMA_F32_16X16X4_F32 (exact f32 matrix core path).
// Edge softmax/aggregation uses ordered-uint atomicMax + f32 atomic adds.
// ---------------------------------------------------------------------------

#define HEADS 4
#define CPH   16
#define HID   64   // HEADS*CPH
#define NEG_SLOPE 0.2f
#define BN_EPS    1e-5f
#define MT    4    // 16-row tiles per wave in the GEMM (64 rows per wave)

typedef __attribute__((ext_vector_type(2))) float v2f;
typedef __attribute__((ext_vector_type(8))) float v8f;

// ------------------------- order-preserving float<->uint -------------------
__device__ __forceinline__ unsigned fenc(float f) {
    unsigned u = __float_as_uint(f);
    return (u & 0x80000000u) ? ~u : (u | 0x80000000u);
}
__device__ __forceinline__ float fdec(unsigned e) {
    return (e & 0x80000000u) ? __uint_as_float(e & 0x7FFFFFFFu)
                             : __uint_as_float(~e);
}
__device__ __forceinline__ float lrelu(float x) {
    return x > 0.0f ? x : NEG_SLOPE * x;
}

// ------------------------- WMMA f32 GEMM: C[M x 64] = A[M x K] @ W[K x 64] --
// One block = 4 waves; each wave owns one 16-col tile of a 64-row M strip
// (MT=4 accumulator tiles, so each B fragment feeds 4 wmma).
// A fragment (16x4 f32): lane&15 = row, (lane>>4)*2 = k offset, 2 floats/lane
//   -> the 2 elements are K-contiguous: one b64 load.
// B fragment (4x16 f32): lane&15 = col, (lane>>4)*2 = k offset (mirror of A).
// D (16x16 f32, 8 VGPRs): vgpr v -> row v + 8*(lane>>4), col = lane&15.
// Out-of-range rows: clamp the load address (row r of D depends only on row r
// of A, so garbage rows are computed but never stored) -> no divergent loads.
__global__ __launch_bounds__(128) void gemm_wmma_f32(
    const float* __restrict__ A, const float* __restrict__ W,
    float* __restrict__ C, int M, int K) {
    const int lane = threadIdx.x & 31;
    const int wave = threadIdx.x >> 5;          // 0..3 -> col tile
    const int row0 = blockIdx.x * (16 * MT);
    const int col0 = wave * 16;
    const int kk   = (lane >> 4) * 2;           // 0 or 2
    const int bcol = col0 + (lane & 15);

    size_t arow[MT];
#pragma unroll
    for (int m = 0; m < MT; ++m) {
        int r = row0 + m * 16 + (lane & 15);
        arow[m] = (size_t)(r < M ? r : M - 1);  // clamped: branchless loads
    }

    v8f acc[MT];
#pragma unroll
    for (int m = 0; m < MT; ++m) acc[m] = (v8f){};

    for (int k0 = 0; k0 < K; k0 += 4) {
        v2f b;
        b.x = W[(size_t)(k0 + kk)     * HID + bcol];
        b.y = W[(size_t)(k0 + kk + 1) * HID + bcol];
#pragma unroll
        for (int m = 0; m < MT; ++m) {
            v2f a = *(const v2f*)(A + arow[m] * K + (k0 + kk));
            acc[m] = __builtin_amdgcn_wmma_f32_16x16x4_f32(
                /*neg_a=*/false, a, /*neg_b=*/false, b,
                /*c_mod=*/(short)0, acc[m],
                /*reuse_a=*/false, /*reuse_b=*/false);
        }
    }

#pragma unroll
    for (int m = 0; m < MT; ++m) {
#pragma unroll
        for (int v = 0; v < 8; ++v) {
            int r = row0 + m * 16 + v + (lane >> 4) * 8;
            if (r < M) C[(size_t)r * HID + bcol] = acc[m][v];
        }
    }
}

// ------------------------- per-node attention logits ------------------------
// als[n,h] = <hp[n,h,:], a_src[h,:]> ; ald likewise.
__global__ void node_alpha(const float* __restrict__ hp,
                           const float* __restrict__ a_s,
                           const float* __restrict__ a_d,
                           float* __restrict__ als, float* __restrict__ ald,
                           int Nn) {
    int t = blockIdx.x * blockDim.x + threadIdx.x;
    if (t >= Nn * HEADS) return;
    int n = t >> 2, h = t & 3;
    const float* hv = hp + (size_t)n * HID + h * CPH;
    float s1 = 0.f, s2 = 0.f;
#pragma unroll
    for (int c = 0; c < CPH; ++c) {
        float v = hv[c];
        s1 += v * a_s[h * CPH + c];
        s2 += v * a_d[h * CPH + c];
    }
    als[t] = s1;
    ald[t] = s2;
}

// ------------------------- fills ------------------------------------------
__global__ void fill_u32(unsigned* __restrict__ p, unsigned v, long long n) {
    long long t = (long long)blockIdx.x * blockDim.x + threadIdx.x;
    if (t < n) p[t] = v;
}
__global__ void init_bias(float* __restrict__ out, const float* __restrict__ b,
                          long long n64) {
    long long t = (long long)blockIdx.x * blockDim.x + threadIdx.x;
    if (t < n64) out[t] = b[t & (HID - 1)];
}

// ------------------------- edge kernels -----------------------------------
__device__ __forceinline__ void edge_endpoints(const int* ei, int E, int idx,
                                               int& src, int& dst) {
    if (idx < E) { src = ei[idx]; dst = ei[E + idx]; }
    else         { src = dst = idx - E; }           // self-loops
}

__global__ void edge_max(const int* __restrict__ ei, int E, int Nn,
                         const float* __restrict__ als,
                         const float* __restrict__ ald,
                         unsigned* __restrict__ menc) {
    int idx = blockIdx.x * blockDim.x + threadIdx.x;
    if (idx >= E + Nn) return;
    int src, dst; edge_endpoints(ei, E, idx, src, dst);
#pragma unroll
    for (int h = 0; h < HEADS; ++h) {
        float e = lrelu(als[src * HEADS + h] + ald[dst * HEADS + h]);
        atomicMax(&menc[dst * HEADS + h], fenc(e));
    }
}

__global__ void edge_sum(const int* __restrict__ ei, int E, int Nn,
                         const float* __restrict__ als,
                         const float* __restrict__ ald,
                         const unsigned* __restrict__ menc,
                         float* __restrict__ ssum) {
    int idx = blockIdx.x * blockDim.x + threadIdx.x;
    if (idx >= E + Nn) return;
    int src, dst; edge_endpoints(ei, E, idx, src, dst);
#pragma unroll
    for (int h = 0; h < HEADS; ++h) {
        float e = lrelu(als[src * HEADS + h] + ald[dst * HEADS + h]);
        float ex = expf(e - fdec(menc[dst * HEADS + h]));
        atomicAdd(&ssum[dst * HEADS + h], ex);
    }
}

// Thread handles one (edge, 8-channel chunk). 8 channels stay inside one head.
__global__ void edge_msg(const int* __restrict__ ei, int E, int Nn,
                         const float* __restrict__ als,
                         const float* __restrict__ ald,
                         const unsigned* __restrict__ menc,
                         const float* __restrict__ ssum,
                         const float* __restrict__ hp,
                         float* __restrict__ out) {
    long long t = (long long)blockIdx.x * blockDim.x + threadIdx.x;
    long long total = (long long)(E + Nn) * 8;
    if (t >= total) return;
    int idx  = (int)(t >> 3);
    int part = (int)(t & 7);          // 8 channels per part
    int src, dst; edge_endpoints(ei, E, idx, src, dst);
    int h = part >> 1;                // head of this channel chunk
    float e = lrelu(als[src * HEADS + h] + ald[dst * HEADS + h]);
    float alpha = expf(e - fdec(menc[dst * HEADS + h])) /
                  (ssum[dst * HEADS + h] + 1e-16f);
    const float4* hv = (const float4*)(hp + (size_t)src * HID + part * 8);
    float4 v0 = hv[0], v1 = hv[1];
    float* ob = out + (size_t)dst * HID + part * 8;
    atomicAdd(ob + 0, v0.x * alpha);
    atomicAdd(ob + 1, v0.y * alpha);
    atomicAdd(ob + 2, v0.z * alpha);
    atomicAdd(ob + 3, v0.w * alpha);
    atomicAdd(ob + 4, v1.x * alpha);
    atomicAdd(ob + 5, v1.y * alpha);
    atomicAdd(ob + 6, v1.z * alpha);
    atomicAdd(ob + 7, v1.w * alpha);
}

// ------------------------- batch norm -------------------------------------
__global__ void bn_stats(const float* __restrict__ x, int Nn,
                         float* __restrict__ stats /*[128]: sum, sumsq*/) {
    __shared__ float ssum[HID], ssq[HID];
    if (threadIdx.x < HID) { ssum[threadIdx.x] = 0.f; ssq[threadIdx.x] = 0.f; }
    __syncthreads();
    int c = threadIdx.x & (HID - 1);
    int rstart  = blockIdx.x * (blockDim.x >> 6) + (threadIdx.x >> 6);
    int rstride = gridDim.x * (blockDim.x >> 6);
    float ls = 0.f, lq = 0.f;
    for (int r = rstart; r < Nn; r += rstride) {
        float v = x[(size_t)r * HID + c];
        ls += v; lq += v * v;
    }
    atomicAdd(&ssum[c], ls);
    atomicAdd(&ssq[c], lq);
    __syncthreads();
    if (threadIdx.x < HID) {
        atomicAdd(&stats[threadIdx.x], ssum[threadIdx.x]);
        atomicAdd(&stats[HID + threadIdx.x], ssq[threadIdx.x]);
    }
}

__global__ void bn_finalize(const float* __restrict__ stats,
                            const float* __restrict__ g,
                            const float* __restrict__ be, int Nn,
                            float* __restrict__ scsh /*[128]: scale, shift*/) {
    int c = threadIdx.x;
    if (c >= HID) return;
    float inv = 1.0f / (float)Nn;
    float mu  = stats[c] * inv;
    float var = stats[HID + c] * inv - mu * mu;
    float sc  = g[c] * rsqrtf(var + BN_EPS);
    scsh[c]       = sc;
    scsh[HID + c] = be[c] - mu * sc;
}

__global__ void bn_apply(const float* __restrict__ x,
                         const float* __restrict__ scsh,
                         float* __restrict__ y, long long n64, int relu) {
    long long t = (long long)blockIdx.x * blockDim.x + threadIdx.x;
    if (t >= n64) return;
    int c = (int)(t & (HID - 1));
    float v = x[t] * scsh[c] + scsh[HID + c];
    if (relu) v = v > 0.f ? v : 0.f;
    y[t] = v;
}

// ---------------------------------------------------------------------------
extern "C" void kernel_launch(void* const* d_in, const int* in_sizes, int n_in,
                              void* d_out, int out_size, void* d_ws, size_t ws_size,
                              hipStream_t stream) {
    const float* x      = (const float*)d_in[0];
    const int*   ei     = (const int*)  d_in[1];
    const float* W0     = (const float*)d_in[2];
    const float* a_src0 = (const float*)d_in[3];
    const float* a_dst0 = (const float*)d_in[4];
    const float* b0     = (const float*)d_in[5];
    const float* g0     = (const float*)d_in[6];
    const float* be0    = (const float*)d_in[7];
    const float* W1     = (const float*)d_in[8];
    const float* a_src1 = (const float*)d_in[9];
    const float* a_dst1 = (const float*)d_in[10];
    const float* b1     = (const float*)d_in[11];
    const float* g1     = (const float*)d_in[12];
    const float* be1    = (const float*)d_in[13];

    const int IN_DIM = 128;
    const int N  = in_sizes[0] / IN_DIM;
    const int E  = in_sizes[1] / 2;
    const int Et = E + N;

    // workspace layout (floats)
    float*    ws    = (float*)d_ws;
    size_t    nf    = (size_t)N;
    float*    hp    = ws;                         // N*64  (hp / hrelu / accum1)
    float*    out   = hp  + nf * HID;             // N*64  (accum0 / hp1)
    float*    als   = out + nf * HID;             // N*4
    float*    ald   = als + nf * HEADS;           // N*4
    unsigned* menc  = (unsigned*)(ald + nf * HEADS); // N*4
    float*    ssum  = (float*)(menc + nf * HEADS);   // N*4
    float*    stats = ssum + nf * HEADS;          // 128
    float*    scsh  = stats + 2 * HID;            // 128
    float*    yout  = (float*)d_out;

    const int T = 256;
    dim3 blk(T);
    int gN4  = (N * HEADS + T - 1) / T;
    int gN64 = (int)(((long long)N * HID + T - 1) / T);
    int gEt  = (Et + T - 1) / T;
    int gMsg = (int)(((long long)Et * 8 + T - 1) / T);
    int gGem = (N + 16 * MT - 1) / (16 * MT);

    // ===================== Layer 0 =====================
    gemm_wmma_f32<<<gGem, 128, 0, stream>>>(x, W0, hp, N, IN_DIM);
    node_alpha<<<gN4, blk, 0, stream>>>(hp, a_src0, a_dst0, als, ald, N);
    fill_u32<<<gN4, blk, 0, stream>>>(menc, 0u, (long long)N * HEADS);
    fill_u32<<<gN4, blk, 0, stream>>>((unsigned*)ssum, 0u, (long long)N * HEADS);
    init_bias<<<gN64, blk, 0, stream>>>(out, b0, (long long)N * HID);
    edge_max<<<gEt, blk, 0, stream>>>(ei, E, N, als, ald, menc);
    edge_sum<<<gEt, blk, 0, stream>>>(ei, E, N, als, ald, menc, ssum);
    edge_msg<<<gMsg, blk, 0, stream>>>(ei, E, N, als, ald, menc, ssum, hp, out);
    fill_u32<<<1, 128, 0, stream>>>((unsigned*)stats, 0u, 2 * HID);
    bn_stats<<<128, blk, 0, stream>>>(out, N, stats);
    bn_finalize<<<1, 64, 0, stream>>>(stats, g0, be0, N, scsh);
    bn_apply<<<gN64, blk, 0, stream>>>(out, scsh, hp, (long long)N * HID, 1);

    // ===================== Layer 1 =====================
    gemm_wmma_f32<<<gGem, 128, 0, stream>>>(hp, W1, out, N, HID);
    node_alpha<<<gN4, blk, 0, stream>>>(out, a_src1, a_dst1, als, ald, N);
    fill_u32<<<gN4, blk, 0, stream>>>(menc, 0u, (long long)N * HEADS);
    fill_u32<<<gN4, blk, 0, stream>>>((unsigned*)ssum, 0u, (long long)N * HEADS);
    init_bias<<<gN64, blk, 0, stream>>>(hp, b1, (long long)N * HID);
    edge_max<<<gEt, blk, 0, stream>>>(ei, E, N, als, ald, menc);
    edge_sum<<<gEt, blk, 0, stream>>>(ei, E, N, als, ald, menc, ssum);
    edge_msg<<<gMsg, blk, 0, stream>>>(ei, E, N, als, ald, menc, ssum, out, hp);
    fill_u32<<<1, 128, 0, stream>>>((unsigned*)stats, 0u, 2 * HID);
    bn_stats<<<128, blk, 0, stream>>>(hp, N, stats);
    bn_finalize<<<1, 64, 0, stream>>>(stats, g1, be1, N, scsh);
    bn_apply<<<gN64, blk, 0, stream>>>(hp, scsh, yout, (long long)N * HID, 0);
}